// MemoryEfficientContrastiveLoss_13821204759229
// MI455X (gfx1250) — compile-verified
//
#include <hip/hip_runtime.h>
#include <math.h>

// -----------------------------------------------------------------------------
// Supervised contrastive loss, MI455X (gfx1250, wave32).
//
//   sim = (F F^T)/T,  T = 0.07,  F: [B,128] f32, B = 8192.
//
// Strategy: never materialize sim (256MB). Each wave owns a 16-row tile and
// streams over all B/16 column tiles twice with V_WMMA_F32_16X16X4_F32
// (exact f32 math, matches the f32 reference):
//   pass 1: row max of the raw dot products
//   pass 2: sum exp((dot - rowmax)/T)  [diagonal included, as in reference],
//           positive-sum and positive-count under the label mask (diag excl.)
// F is 4MB -> L2 resident; A fragments (16x128 f32) live in 64 VGPRs/lane and
// are reused across all 512 column tiles, so the inner loop is 32 WMMAs + 32
// float2 loads per tile.
// -----------------------------------------------------------------------------

typedef float v2f __attribute__((ext_vector_type(2)));
typedef float v8f __attribute__((ext_vector_type(8)));

#define INV_TEMP (1.0f / 0.07f)
#define DIM 128

__global__ __launch_bounds__(64) void supcon_rowtiles_kernel(
    const float* __restrict__ F,
    const int*   __restrict__ lab,
    float*       __restrict__ ws,   // ws[0] = -sum(mlp*valid), ws[1] = n_valid
    int B)
{
    const int lane     = threadIdx.x & 31;
    const int waveInBlk= threadIdx.x >> 5;
    const int rowTile  = blockIdx.x * (blockDim.x >> 5) + waveInBlk;
    const int nTiles   = B >> 4;                 // 16 rows/cols per tile
    if (rowTile >= nTiles) return;               // whole-wave exit: EXEC stays all-1s

    const int r0   = rowTile << 4;
    const int half = lane >> 4;                  // 0: lanes 0-15, 1: lanes 16-31
    const int l15  = lane & 15;

    // ---- A fragments: 32-bit A 16x4 layout. lane -> M=l15, K0 = 2*half.
    // frag t covers K = 4t .. 4t+3; this lane holds (K=4t+2*half, 4t+2*half+1),
    // contiguous in row-major F -> one float2 load each. 64 VGPRs total.
    const float* aBase = F + (size_t)(r0 + l15) * DIM + 2 * half;
    v2f aFrag[32];
#pragma unroll
    for (int t = 0; t < 32; ++t)
        aFrag[t] = *(const v2f*)(aBase + 4 * t);

    // Labels for this lane's 8 rows (C/D layout: row = r0 + 8*half + v).
    int labRow[8];
#pragma unroll
    for (int v = 0; v < 8; ++v)
        labRow[v] = lab[r0 + 8 * half + v];

    // =========================== pass 1: row max ===========================
    float dmax[8];
#pragma unroll
    for (int v = 0; v < 8; ++v) dmax[v] = -3.4e38f;

    for (int ct = 0; ct < nTiles; ++ct) {
        // B fragment: 32-bit B 4x16 layout. lane -> N=l15, K0 = 2*half.
        const float* bBase = F + (size_t)(ct * 16 + l15) * DIM + 2 * half;
        v8f acc = {};
#pragma unroll
        for (int t = 0; t < 32; ++t) {
            v2f b = *(const v2f*)(bBase + 4 * t);
            acc = __builtin_amdgcn_wmma_f32_16x16x4_f32(
                false, aFrag[t], false, b, (short)0, acc, false, false);
        }
#pragma unroll
        for (int v = 0; v < 8; ++v) dmax[v] = fmaxf(dmax[v], acc[v]);
    }
    // Reduce max across the 16 lanes of each half (masks < 16 stay in-half).
#pragma unroll
    for (int m = 1; m < 16; m <<= 1) {
#pragma unroll
        for (int v = 0; v < 8; ++v)
            dmax[v] = fmaxf(dmax[v], __shfl_xor(dmax[v], m, 32));
    }

    // ================= pass 2: sum-exp, positive sum/count =================
    float S[8], P[8], C[8];
#pragma unroll
    for (int v = 0; v < 8; ++v) { S[v] = 0.0f; P[v] = 0.0f; C[v] = 0.0f; }

    for (int ct = 0; ct < nTiles; ++ct) {
        const int col  = ct * 16 + l15;
        const int labC = lab[col];
        const float* bBase = F + (size_t)col * DIM + 2 * half;
        v8f acc = {};
#pragma unroll
        for (int t = 0; t < 32; ++t) {
            v2f b = *(const v2f*)(bBase + 4 * t);
            acc = __builtin_amdgcn_wmma_f32_16x16x4_f32(
                false, aFrag[t], false, b, (short)0, acc, false, false);
        }
#pragma unroll
        for (int v = 0; v < 8; ++v) {
            const float x   = acc[v];
            const int   row = r0 + 8 * half + v;
            // denominator includes the diagonal term (as in the reference)
            S[v] += __expf((x - dmax[v]) * INV_TEMP);
            const bool pos = (labC == labRow[v]) && (col != row);
            P[v] += pos ? x : 0.0f;
            C[v] += pos ? 1.0f : 0.0f;
        }
    }
    // Sum-reduce S, P, C across each 16-lane half.
#pragma unroll
    for (int m = 1; m < 16; m <<= 1) {
#pragma unroll
        for (int v = 0; v < 8; ++v) {
            S[v] += __shfl_xor(S[v], m, 32);
            P[v] += __shfl_xor(P[v], m, 32);
            C[v] += __shfl_xor(C[v], m, 32);
        }
    }

    // ===================== per-row loss, global accumulate =================
    if (l15 == 0) {   // lane 0 handles rows r0..r0+7, lane 16 rows r0+8..r0+15
        float num = 0.0f, nv = 0.0f;
#pragma unroll
        for (int v = 0; v < 8; ++v) {
            const float c = C[v];
            // sum_pos(log_prob) = (P - c*dmax)/T - c*log(S + 1e-8)
            const float mlp = ((P[v] - c * dmax[v]) * INV_TEMP
                               - c * __logf(S[v] + 1e-8f)) / (c + 1e-8f);
            if (c > 0.0f) { num -= mlp; nv += 1.0f; }
        }
        atomicAdd(&ws[0], num);
        atomicAdd(&ws[1], nv);
    }
}

__global__ void supcon_finalize_kernel(const float* __restrict__ ws,
                                       float* __restrict__ out)
{
    const float nv = ws[1];
    out[0] = (nv > 0.0f) ? (ws[0] / fmaxf(nv, 1.0f)) : 0.0f;
}

extern "C" void kernel_launch(void* const* d_in, const int* in_sizes, int n_in,
                              void* d_out, int out_size, void* d_ws, size_t ws_size,
                              hipStream_t stream)
{
    const float* F   = (const float*)d_in[0];
    const int*   lab = (const int*)d_in[1];
    float*       out = (float*)d_out;
    float*       ws  = (float*)d_ws;

    const int B      = in_sizes[0] / DIM;   // 8192
    const int nTiles = B >> 4;              // 512 row tiles, one wave each

    // Zero the two accumulators every call (graph-capture safe).
    hipMemsetAsync(ws, 0, 2 * sizeof(float), stream);

    const int threads = 64;                                  // 2 waves / block
    const int wavesPerBlock = threads / 32;
    const int blocks = (nTiles + wavesPerBlock - 1) / wavesPerBlock;

    supcon_rowtiles_kernel<<<blocks, threads, 0, stream>>>(F, lab, ws, B);
    supcon_finalize_kernel<<<1, 1, 0, stream>>>(ws, out);
}